// SS2D_24240795418819
// MI455X (gfx1250) — compile-verified
//
#include <hip/hip_runtime.h>
#include <hip/hip_bf16.h>

typedef float v2f __attribute__((ext_vector_type(2)));
typedef float v8f __attribute__((ext_vector_type(8)));

#define D_MODEL 96
#define D_INNER 192
#define D_STATE 16
#define DT_RANK 6
#define KDIR 4
#define BATCH 8
#define HH 48
#define WW 48
#define LL 2304              // HH*WW
#define XDBL_STRIDE 40       // 38 rounded up -> chunk rows are contiguous
#define CHUNK 128            // scan steps staged per LDS tile
#define NCHUNK (LL / CHUNK)  // 18

__device__ __forceinline__ v8f wmma_f32(v2f a, v2f b, v8f c) {
    return __builtin_amdgcn_wmma_f32_16x16x4_f32(false, a, false, b,
                                                 (short)0, c, false, false);
}

// async global->LDS copy, 16 bytes per active lane (ASYNCcnt-tracked)
__device__ __forceinline__ void async_copy_b128(unsigned lds_byte_off, const void* gaddr) {
    asm volatile("global_load_async_to_lds_b128 %0, %1, off"
                 :: "v"(lds_byte_off), "v"(gaddr) : "memory");
}
__device__ __forceinline__ void wait_asynccnt0() {
    asm volatile("s_wait_asynccnt 0x0" ::: "memory");
}

// scan-order permutation (all four are involutions: gather index == scatter index)
__device__ __forceinline__ int perm_idx(int k, int l) {
    if (k == 0) return l;
    if (k == 1) return (l % WW) * WW + (l / WW);
    if (k == 2) return LL - 1 - l;
    int lr = LL - 1 - l;
    return (lr % WW) * WW + (lr / WW);
}

// ---------------------------------------------------------------------------
// K1: in_proj  h0[b,o,l] = sum_c W[o,c] * x[b,c,l]   (M=192,K=96,N=2304 per b)
// one wave per 16x64 tile: 4 accumulators share each A fetch (fp32 WMMA)
// ---------------------------------------------------------------------------
__global__ void k_in_proj(const float* __restrict__ x, const float* __restrict__ W,
                          float* __restrict__ h0) {
    int lane = threadIdx.x;
    bool hi  = lane >= 16;
    int  lx  = lane & 15;
    int n0 = blockIdx.x * 64;
    int m0 = blockIdx.y * 16;
    int b  = blockIdx.z;

    const float* Wrow = W + (size_t)(m0 + lx) * D_MODEL;
    const float* xcol = x + (size_t)b * D_MODEL * LL + n0 + lx;

    v8f a0 = {}, a1 = {}, a2 = {}, a3 = {};
    for (int kk = 0; kk < D_MODEL; kk += 4) {
        int kb = kk + (hi ? 2 : 0);
        v2f a;  a.x = Wrow[kb]; a.y = Wrow[kb + 1];
        const float* c0 = xcol + (size_t)kb * LL;
        const float* c1 = xcol + (size_t)(kb + 1) * LL;
        v2f b0; b0.x = c0[0];  b0.y = c1[0];
        v2f b1; b1.x = c0[16]; b1.y = c1[16];
        v2f b2; b2.x = c0[32]; b2.y = c1[32];
        v2f b3; b3.x = c0[48]; b3.y = c1[48];
        a0 = wmma_f32(a, b0, a0);
        a1 = wmma_f32(a, b1, a1);
        a2 = wmma_f32(a, b2, a2);
        a3 = wmma_f32(a, b3, a3);
    }
    float* drow = h0 + ((size_t)b * D_INNER + m0 + (hi ? 8 : 0)) * LL + n0 + lx;
    #pragma unroll
    for (int v = 0; v < 8; ++v) {
        drow[(size_t)v * LL + 0]  = a0[v];
        drow[(size_t)v * LL + 16] = a1[v];
        drow[(size_t)v * LL + 32] = a2[v];
        drow[(size_t)v * LL + 48] = a3[v];
    }
}

// ---------------------------------------------------------------------------
// K2: depthwise 3x3 conv + bias + SiLU
// ---------------------------------------------------------------------------
__global__ void k_conv_silu(const float* __restrict__ h0, const float* __restrict__ cw,
                            const float* __restrict__ cb, float* __restrict__ hs) {
    int idx = blockIdx.x * blockDim.x + threadIdx.x;
    const int total = BATCH * D_INNER * LL;
    if (idx >= total) return;
    int xw = idx % WW;
    int yh = (idx / WW) % HH;
    int c  = (idx / LL) % D_INNER;
    int b  = idx / (D_INNER * LL);

    const float* base = h0 + ((size_t)b * D_INNER + c) * LL;
    const float* wv   = cw + (size_t)c * 9;
    float acc = cb[c];
    #pragma unroll
    for (int dy = 0; dy < 3; ++dy) {
        int iy = yh + dy - 1;
        if (iy < 0 || iy >= HH) continue;
        #pragma unroll
        for (int dx = 0; dx < 3; ++dx) {
            int ix = xw + dx - 1;
            if (ix < 0 || ix >= WW) continue;
            acc += wv[dy * 3 + dx] * base[iy * WW + ix];
        }
    }
    hs[idx] = acc / (1.0f + __expf(-acc));   // SiLU
}

// ---------------------------------------------------------------------------
// K3: x_proj  x_dbl[b,k,l,o] = sum_d Wp[k,o,d] * hs[b,d,perm_k(l)]
// M=38 (3 row tiles, masked), K=192, N tiles of 64. Stored [b,k,l,40].
// ---------------------------------------------------------------------------
__global__ void k_xproj(const float* __restrict__ hs, const float* __restrict__ Wp,
                        float* __restrict__ xdbl) {
    int lane = threadIdx.x;
    bool hi  = lane >= 16;
    int  lx  = lane & 15;
    int n0 = blockIdx.x * 64;
    int m0 = blockIdx.y * 16;
    int bk = blockIdx.z;
    int b  = bk >> 2;
    int k  = bk & 3;

    int  m   = m0 + lx;
    bool mok = m < (DT_RANK + 2 * D_STATE);

    const float* Wrow = Wp + ((size_t)k * (DT_RANK + 2 * D_STATE) + m) * D_INNER;
    const float* hbb  = hs + (size_t)b * D_INNER * LL;
    int pl0 = perm_idx(k, n0 + 0  + lx);
    int pl1 = perm_idx(k, n0 + 16 + lx);
    int pl2 = perm_idx(k, n0 + 32 + lx);
    int pl3 = perm_idx(k, n0 + 48 + lx);

    v8f a0 = {}, a1 = {}, a2 = {}, a3 = {};
    for (int kk = 0; kk < D_INNER; kk += 4) {
        int kb = kk + (hi ? 2 : 0);
        v2f a;  a.x = mok ? Wrow[kb] : 0.0f;  a.y = mok ? Wrow[kb + 1] : 0.0f;
        const float* r0 = hbb + (size_t)kb * LL;
        const float* r1 = hbb + (size_t)(kb + 1) * LL;
        v2f b0; b0.x = r0[pl0]; b0.y = r1[pl0];
        v2f b1; b1.x = r0[pl1]; b1.y = r1[pl1];
        v2f b2; b2.x = r0[pl2]; b2.y = r1[pl2];
        v2f b3; b3.x = r0[pl3]; b3.y = r1[pl3];
        a0 = wmma_f32(a, b0, a0);
        a1 = wmma_f32(a, b1, a1);
        a2 = wmma_f32(a, b2, a2);
        a3 = wmma_f32(a, b3, a3);
    }
    int mrow = m0 + (hi ? 8 : 0);
    float* dbase = xdbl + (size_t)bk * LL * XDBL_STRIDE + mrow;
    #pragma unroll
    for (int v = 0; v < 8; ++v) {
        int m2 = mrow + v;
        if (m2 < (DT_RANK + 2 * D_STATE)) {
            dbase[(size_t)(n0 + 0  + lx) * XDBL_STRIDE + v] = a0[v];
            dbase[(size_t)(n0 + 16 + lx) * XDBL_STRIDE + v] = a1[v];
            dbase[(size_t)(n0 + 32 + lx) * XDBL_STRIDE + v] = a2[v];
            dbase[(size_t)(n0 + 48 + lx) * XDBL_STRIDE + v] = a3[v];
        }
    }
}

// ---------------------------------------------------------------------------
// K4: zero the accumulation buffer (ws is poisoned by the harness)
// ---------------------------------------------------------------------------
__global__ void k_zero(float* __restrict__ p, int n) {
    int i = blockIdx.x * blockDim.x + threadIdx.x;
    if (i < n) p[i] = 0.0f;
}

// ---------------------------------------------------------------------------
// K5: selective scan with async double-buffered LDS staging.
// Block = 1024 threads = 32 waves = 64 channels of one (b,k,group).
// Each chunk of 128 steps (20 KB of x_dbl rows, contiguous) is DMA'd to LDS
// with global_load_async_to_lds_b128 while the previous chunk's serial
// recurrence runs; dt/B/C reads then come from LDS.
// ---------------------------------------------------------------------------
__global__ void __launch_bounds__(1024)
k_scan(const float* __restrict__ hs, const float* __restrict__ xdbl,
       const float* __restrict__ dtw, const float* __restrict__ dtb,
       const float* __restrict__ A_logs, const float* __restrict__ Ds_,
       float* __restrict__ y) {
    __shared__ float sbuf[2 * CHUNK * XDBL_STRIDE];   // 40 KB, LDS offset 0

    int blk = blockIdx.x;                 // 0..95
    int g   = blk % 3;                    // channel group (64 channels each)
    int k   = (blk / 3) % KDIR;
    int b   = blk / (3 * KDIR);

    int tid  = threadIdx.x;
    int wave = tid >> 5;                  // 0..31
    int lane = tid & 31;
    bool hi  = lane >= 16;
    int  s   = lane & 15;                 // state index
    int  d   = (g * 32 + wave) * 2 + (hi ? 1 : 0);
    int  kd  = k * D_INNER + d;

    float Aval = -__expf(A_logs[(size_t)kd * D_STATE + s]);
    float w0 = dtw[kd * DT_RANK + 0], w1 = dtw[kd * DT_RANK + 1];
    float w2 = dtw[kd * DT_RANK + 2], w3 = dtw[kd * DT_RANK + 3];
    float w4 = dtw[kd * DT_RANK + 4], w5 = dtw[kd * DT_RANK + 5];
    float bias = dtb[kd];
    float Dsv  = Ds_[kd];

    const float* hrow = hs + ((size_t)b * D_INNER + d) * LL;
    float*       yrow = y  + ((size_t)b * D_INNER + d) * LL;
    const float* xd   = xdbl + (size_t)(b * KDIR + k) * LL * XDBL_STRIDE;

    // stage chunk 0 into buffer 0 (1280 x b128 over 1024 threads)
    {
        const float* src = xd;
        if (tid < CHUNK * XDBL_STRIDE / 4)
            async_copy_b128((unsigned)(tid * 16), src + tid * 4);
        int i2 = tid + 1024;
        if (i2 < CHUNK * XDBL_STRIDE / 4)
            async_copy_b128((unsigned)(i2 * 16), src + i2 * 4);
    }

    float hst = 0.0f;
    for (int c = 0; c < NCHUNK; ++c) {
        wait_asynccnt0();
        __syncthreads();                 // chunk c resident for all waves

        if (c + 1 < NCHUNK) {            // overlap: DMA next chunk
            unsigned base = (unsigned)(((c + 1) & 1) * CHUNK * XDBL_STRIDE * 4);
            const float* src = xd + (size_t)(c + 1) * CHUNK * XDBL_STRIDE;
            if (tid < CHUNK * XDBL_STRIDE / 4)
                async_copy_b128(base + tid * 16, src + tid * 4);
            int i2 = tid + 1024;
            if (i2 < CHUNK * XDBL_STRIDE / 4)
                async_copy_b128(base + i2 * 16, src + i2 * 4);
        }

        const float* lrow = sbuf + (c & 1) * CHUNK * XDBL_STRIDE;
        for (int r = 0; r < CHUNK; ++r) {
            const float* row = lrow + r * XDBL_STRIDE;      // LDS reads
            float dtraw = bias + w0 * row[0] + w1 * row[1] + w2 * row[2]
                               + w3 * row[3] + w4 * row[4] + w5 * row[5];
            float dtv = dtraw > 20.0f ? dtraw : log1pf(__expf(dtraw));  // softplus
            float Bv = row[DT_RANK + s];
            float Cv = row[DT_RANK + D_STATE + s];

            int l  = c * CHUNK + r;
            int pl = perm_idx(k, l);
            float u = hrow[pl];

            hst = hst * __expf(dtv * Aval) + dtv * u * Bv;
            float prod = hst * Cv;
            prod += __shfl_xor(prod, 1, 32);
            prod += __shfl_xor(prod, 2, 32);
            prod += __shfl_xor(prod, 4, 32);
            prod += __shfl_xor(prod, 8, 32);
            if (s == 0) atomicAdd(&yrow[pl], prod + Dsv * u);
        }
        __syncthreads();                 // done reading buf[c&1] before re-stage
    }
}

// ---------------------------------------------------------------------------
// K6: LayerNorm over channels (192) per (b,l); one wave per position
// ---------------------------------------------------------------------------
__global__ void k_layernorm(const float* __restrict__ y, const float* __restrict__ g,
                            const float* __restrict__ bta, float* __restrict__ yn) {
    int pos = blockIdx.x;                      // 0 .. B*LL-1
    int b = pos / LL;
    int l = pos % LL;
    int lane = threadIdx.x;

    const float* col = y + (size_t)b * D_INNER * LL + l;
    float v[6];
    float sum = 0.0f, sq = 0.0f;
    #pragma unroll
    for (int j = 0; j < 6; ++j) {
        float t = col[(size_t)(lane + j * 32) * LL];
        v[j] = t; sum += t; sq += t * t;
    }
    #pragma unroll
    for (int off = 16; off > 0; off >>= 1) {
        sum += __shfl_xor(sum, off, 32);
        sq  += __shfl_xor(sq,  off, 32);
    }
    float mu  = sum * (1.0f / D_INNER);
    float var = sq * (1.0f / D_INNER) - mu * mu;
    float inv = rsqrtf(var + 1e-5f);

    float* oc = yn + (size_t)b * D_INNER * LL + l;
    #pragma unroll
    for (int j = 0; j < 6; ++j) {
        int dch = lane + j * 32;
        oc[(size_t)dch * LL] = (v[j] - mu) * inv * g[dch] + bta[dch];
    }
}

// ---------------------------------------------------------------------------
// K7: out_proj  out[b,o,l] = sum_c W2[o,c] * yn[b,c,l]  (M=96,K=192,N=2304)
// 16x64 tile per wave
// ---------------------------------------------------------------------------
__global__ void k_out_proj(const float* __restrict__ yn, const float* __restrict__ W2,
                           float* __restrict__ out) {
    int lane = threadIdx.x;
    bool hi  = lane >= 16;
    int  lx  = lane & 15;
    int n0 = blockIdx.x * 64;
    int m0 = blockIdx.y * 16;
    int b  = blockIdx.z;

    const float* Wrow = W2 + (size_t)(m0 + lx) * D_INNER;
    const float* ycol = yn + (size_t)b * D_INNER * LL + n0 + lx;

    v8f a0 = {}, a1 = {}, a2 = {}, a3 = {};
    for (int kk = 0; kk < D_INNER; kk += 4) {
        int kb = kk + (hi ? 2 : 0);
        v2f a;  a.x = Wrow[kb]; a.y = Wrow[kb + 1];
        const float* c0 = ycol + (size_t)kb * LL;
        const float* c1 = ycol + (size_t)(kb + 1) * LL;
        v2f b0; b0.x = c0[0];  b0.y = c1[0];
        v2f b1; b1.x = c0[16]; b1.y = c1[16];
        v2f b2; b2.x = c0[32]; b2.y = c1[32];
        v2f b3; b3.x = c0[48]; b3.y = c1[48];
        a0 = wmma_f32(a, b0, a0);
        a1 = wmma_f32(a, b1, a1);
        a2 = wmma_f32(a, b2, a2);
        a3 = wmma_f32(a, b3, a3);
    }
    float* drow = out + ((size_t)b * D_MODEL + m0 + (hi ? 8 : 0)) * LL + n0 + lx;
    #pragma unroll
    for (int v = 0; v < 8; ++v) {
        drow[(size_t)v * LL + 0]  = a0[v];
        drow[(size_t)v * LL + 16] = a1[v];
        drow[(size_t)v * LL + 32] = a2[v];
        drow[(size_t)v * LL + 48] = a3[v];
    }
}

// ---------------------------------------------------------------------------
extern "C" void kernel_launch(void* const* d_in, const int* in_sizes, int n_in,
                              void* d_out, int out_size, void* d_ws, size_t ws_size,
                              hipStream_t stream) {
    const float* x          = (const float*)d_in[0];
    const float* in_proj_w  = (const float*)d_in[1];
    const float* conv_w     = (const float*)d_in[2];
    const float* conv_b     = (const float*)d_in[3];
    const float* x_proj_w   = (const float*)d_in[4];
    const float* dt_projs_w = (const float*)d_in[5];
    const float* dt_projs_b = (const float*)d_in[6];
    const float* A_logs     = (const float*)d_in[7];
    const float* Ds_        = (const float*)d_in[8];
    const float* norm_g     = (const float*)d_in[9];
    const float* norm_b     = (const float*)d_in[10];
    const float* out_proj_w = (const float*)d_in[11];
    float* out = (float*)d_out;

    const size_t N_ACT  = (size_t)BATCH * D_INNER * LL;                 // 3,538,944
    const size_t N_XDBL = (size_t)BATCH * KDIR * LL * XDBL_STRIDE;      // 2,949,120

    float* ws   = (float*)d_ws;
    float* h0   = ws;                // in_proj output
    float* hs   = h0   + N_ACT;     // conv+silu output
    float* xdbl = hs   + N_ACT;     // [b,k,l,40]
    float* yb   = xdbl + N_XDBL;    // scan accumulation
    float* yn   = yb   + N_ACT;     // normalized

    // 1) in_proj GEMM (WMMA, 16x64 tiles)
    k_in_proj<<<dim3(LL / 64, D_INNER / 16, BATCH), 32, 0, stream>>>(x, in_proj_w, h0);
    // 2) depthwise conv + SiLU
    k_conv_silu<<<(int)((N_ACT + 255) / 256), 256, 0, stream>>>(h0, conv_w, conv_b, hs);
    // 3) x_proj GEMM with per-direction permuted gathers (WMMA)
    k_xproj<<<dim3(LL / 64, 3, BATCH * KDIR), 32, 0, stream>>>(hs, x_proj_w, xdbl);
    // 4) zero scan accumulator
    k_zero<<<(int)((N_ACT + 255) / 256), 256, 0, stream>>>(yb, (int)N_ACT);
    // 5) selective scan: async-LDS double buffering + fused dt-proj/softplus
    k_scan<<<BATCH * KDIR * 3, 1024, 0, stream>>>(hs, xdbl, dt_projs_w,
                                                  dt_projs_b, A_logs, Ds_, yb);
    // 6) channel LayerNorm
    k_layernorm<<<BATCH * LL, 32, 0, stream>>>(yb, norm_g, norm_b, yn);
    // 7) out_proj GEMM (WMMA, 16x64 tiles)
    k_out_proj<<<dim3(LL / 64, D_MODEL / 16, BATCH), 32, 0, stream>>>(yn, out_proj_w, out);
}